// AdaptiveSoftmax_74990128988321
// MI455X (gfx1250) — compile-verified
//
#include <hip/hip_runtime.h>
#include <math.h>

typedef __attribute__((ext_vector_type(2))) float v2f;
typedef __attribute__((ext_vector_type(4))) float v4f;
typedef __attribute__((ext_vector_type(8))) float v8f;

#define BK 128
#define LDS_STRIDE (BK + 4)      // pad 4 dwords -> conflict-free ds_load_b64 pattern
#define WAVES_PER_BLOCK 4
#define NSEG 64
#define RED_THREADS 256

// ---------------------------------------------------------------------------
// Order-preserving float<->uint mapping so per-row max can use uint atomicMax
// (exactly associative -> bitwise deterministic across replays).
// ---------------------------------------------------------------------------
__device__ __forceinline__ unsigned fkey(float f) {
    unsigned u = __float_as_uint(f);
    return (u & 0x80000000u) ? ~u : (u | 0x80000000u);
}
__device__ __forceinline__ float funkey(unsigned k) {
    unsigned u = (k & 0x80000000u) ? (k ^ 0x80000000u) : ~k;
    return __uint_as_float(u);
}

// ---------------------------------------------------------------------------
// Kernel 0: init workspace (max keys). Key 0 is below every real value's key.
// ---------------------------------------------------------------------------
__global__ void init_ws_kernel(unsigned* __restrict__ maxkeys,
                               float* __restrict__ sums, int batch) {
    int i = threadIdx.x;
    if (i < batch) { maxkeys[i] = 0u; sums[i] = 0.0f; }
}

// ---------------------------------------------------------------------------
// Kernel 1: logits = x @ A^T using V_WMMA_F32_16X16X4_F32 (fp32, exact).
// Each wave: 32(M) x 16(N) tile; 2 WMMAs per K-step share one B fragment so
// the 512 MB atom matrix is streamed from HBM exactly once (NT hint).
// x is staged per 128-wide K chunk into padded LDS.
// ---------------------------------------------------------------------------
__global__ __launch_bounds__(32 * WAVES_PER_BLOCK)
void gemm_logits_kernel(const float* __restrict__ x,
                        const float* __restrict__ A,
                        float* __restrict__ out,
                        int d, int n_atoms) {
    __shared__ float xs[32 * LDS_STRIDE];

    const int lane = threadIdx.x & 31;
    const int wave = threadIdx.x >> 5;
    const int half = lane >> 4;          // 0: K={0,1}  1: K={2,3}
    const int mrow = lane & 15;          // N index within tile (B/C lanes)

    const int ntile  = blockIdx.x * WAVES_PER_BLOCK + wave;
    const int n0     = ntile * 16;
    const bool active = (n0 + 16 <= n_atoms);  // wave-uniform -> EXEC stays full
    const int nsafe  = active ? n0 : 0;

    // B fragment source: lane (h, n) loads A[n0+n][k + 2h .. +1]  (4x16 K-major)
    const float* bbase = A + (size_t)(nsafe + mrow) * d + 2 * half;

    // A fragment LDS offsets: lane (h, m) reads x[m][k + 2h .. +1]  (16x4)
    const int aoff0 = mrow * LDS_STRIDE + 2 * half;          // M-tile 0 (rows 0..15)
    const int aoff1 = aoff0 + 16 * LDS_STRIDE;               // M-tile 1 (rows 16..31)

    v8f acc0 = {0.f, 0.f, 0.f, 0.f, 0.f, 0.f, 0.f, 0.f};
    v8f acc1 = {0.f, 0.f, 0.f, 0.f, 0.f, 0.f, 0.f, 0.f};

    for (int kc = 0; kc < d; kc += BK) {
        // Cooperative stage of x[0:32][kc:kc+BK] into LDS (16B vectors, padded rows)
        for (int idx = threadIdx.x; idx < 32 * (BK / 4); idx += blockDim.x) {
            int row = idx >> 5;              // BK/4 == 32 float4 per row
            int c4  = (idx & 31) << 2;
            v4f v = *(const v4f*)(x + (size_t)row * d + kc + c4);
            *(v4f*)(&xs[row * LDS_STRIDE + c4]) = v;
        }
        __syncthreads();

        const float* bp = bbase + kc;
        #pragma unroll 8
        for (int kk = 0; kk < BK; kk += 4) {
            // streaming load of the atom matrix: non-temporal (used once)
            v2f b  = __builtin_nontemporal_load((const v2f*)(bp + kk));
            v2f a0 = *(const v2f*)(&xs[aoff0 + kk]);
            v2f a1 = *(const v2f*)(&xs[aoff1 + kk]);
            acc0 = __builtin_amdgcn_wmma_f32_16x16x4_f32(
                false, a0, false, b, (short)0, acc0, false, false);
            acc1 = __builtin_amdgcn_wmma_f32_16x16x4_f32(
                false, a1, false, b, (short)0, acc1, false, false);
        }
        __syncthreads();
    }

    if (active) {
        // C/D layout: VGPR r on lane l -> M = r + 8*(l>=16), N = l%16
        float* o0 = out + (size_t)(half * 8) * n_atoms + n0 + mrow;
        #pragma unroll
        for (int r = 0; r < 8; ++r) {
            o0[(size_t)r * n_atoms]        = acc0[r];
            o0[(size_t)(r + 16) * n_atoms] = acc1[r];
        }
    }
}

// ---------------------------------------------------------------------------
// Kernel 2: per-row max via block reduce + uint atomicMax (deterministic).
// ---------------------------------------------------------------------------
__global__ __launch_bounds__(RED_THREADS)
void row_max_kernel(const float* __restrict__ logits,
                    unsigned* __restrict__ maxkeys, int n_atoms) {
    __shared__ unsigned sm[RED_THREADS];
    const int b = blockIdx.y;
    const int s = blockIdx.x;
    const float* row = logits + (size_t)b * n_atoms;
    const int per = (n_atoms + NSEG - 1) / NSEG;
    int beg = s * per;
    int end = beg + per; if (end > n_atoms) end = n_atoms;

    unsigned k = 0u;
    for (int i = beg + threadIdx.x; i < end; i += RED_THREADS) {
        unsigned kk = fkey(row[i]);
        k = (kk > k) ? kk : k;
    }
    sm[threadIdx.x] = k;
    __syncthreads();
    for (int off = RED_THREADS / 2; off > 0; off >>= 1) {
        if (threadIdx.x < off) {
            unsigned o = sm[threadIdx.x + off];
            if (o > sm[threadIdx.x]) sm[threadIdx.x] = o;
        }
        __syncthreads();
    }
    if (threadIdx.x == 0) atomicMax(&maxkeys[b], sm[0]);
}

// ---------------------------------------------------------------------------
// Kernel 3: deterministic partial exp-sums (no float atomics).
// ---------------------------------------------------------------------------
__global__ __launch_bounds__(RED_THREADS)
void row_sum_partial_kernel(const float* __restrict__ logits,
                            const unsigned* __restrict__ maxkeys,
                            float* __restrict__ partial, int n_atoms) {
    __shared__ float sm[RED_THREADS];
    const int b = blockIdx.y;
    const int s = blockIdx.x;
    const float mx = funkey(maxkeys[b]);
    const float* row = logits + (size_t)b * n_atoms;
    const int per = (n_atoms + NSEG - 1) / NSEG;
    int beg = s * per;
    int end = beg + per; if (end > n_atoms) end = n_atoms;

    float acc = 0.0f;
    for (int i = beg + threadIdx.x; i < end; i += RED_THREADS)
        acc += __expf(row[i] - mx);
    sm[threadIdx.x] = acc;
    __syncthreads();
    for (int off = RED_THREADS / 2; off > 0; off >>= 1) {
        if (threadIdx.x < off) sm[threadIdx.x] += sm[threadIdx.x + off];
        __syncthreads();
    }
    if (threadIdx.x == 0) partial[b * NSEG + s] = sm[0];
}

// ---------------------------------------------------------------------------
// Kernel 4: finalize per-row sums (fixed tree order -> deterministic).
// ---------------------------------------------------------------------------
__global__ __launch_bounds__(NSEG)
void row_sum_final_kernel(const float* __restrict__ partial,
                          float* __restrict__ sums) {
    __shared__ float sm[NSEG];
    const int b = blockIdx.x;
    sm[threadIdx.x] = partial[b * NSEG + threadIdx.x];
    __syncthreads();
    for (int off = NSEG / 2; off > 0; off >>= 1) {
        if (threadIdx.x < off) sm[threadIdx.x] += sm[threadIdx.x + off];
        __syncthreads();
    }
    if (threadIdx.x == 0) sums[b] = sm[0];
}

// ---------------------------------------------------------------------------
// Kernel 5: normalize in place: out = exp(l - max) / sum.
// ---------------------------------------------------------------------------
__global__ __launch_bounds__(256)
void normalize_kernel(float* __restrict__ out,
                      const unsigned* __restrict__ maxkeys,
                      const float* __restrict__ sums, int n_atoms) {
    const int b = blockIdx.y;
    const int i = blockIdx.x * 256 + threadIdx.x;
    if (i < n_atoms) {
        const float mx  = funkey(maxkeys[b]);
        const float inv = 1.0f / sums[b];
        const size_t idx = (size_t)b * n_atoms + i;
        out[idx] = __expf(out[idx] - mx) * inv;
    }
}

// ---------------------------------------------------------------------------
extern "C" void kernel_launch(void* const* d_in, const int* in_sizes, int n_in,
                              void* d_out, int out_size, void* d_ws, size_t ws_size,
                              hipStream_t stream) {
    const float* x = (const float*)d_in[0];       // [32, 1024]
    const float* A = (const float*)d_in[1];       // [128000, 1024]
    float* out = (float*)d_out;                   // [32, 128000]

    const int batch   = 32;                        // gemm kernel assumes M == 32
    const int d       = in_sizes[0] / batch;       // 1024
    const int n_atoms = out_size / batch;          // 128000

    unsigned* maxkeys = (unsigned*)d_ws;           // 32 uints
    float*    sums    = (float*)d_ws + 32;         // 32 floats
    float*    partial = (float*)d_ws + 64;         // 32*NSEG floats (~8 KB)

    init_ws_kernel<<<1, 64, 0, stream>>>(maxkeys, sums, batch);

    const int ntiles  = (n_atoms + 15) / 16;
    const int nblocks = (ntiles + WAVES_PER_BLOCK - 1) / WAVES_PER_BLOCK;
    gemm_logits_kernel<<<nblocks, 32 * WAVES_PER_BLOCK, 0, stream>>>(
        x, A, out, d, n_atoms);

    dim3 rg(NSEG, batch);
    row_max_kernel<<<rg, RED_THREADS, 0, stream>>>(out, maxkeys, n_atoms);
    row_sum_partial_kernel<<<rg, RED_THREADS, 0, stream>>>(out, maxkeys, partial, n_atoms);
    row_sum_final_kernel<<<batch, NSEG, 0, stream>>>(partial, sums);

    dim3 ng((n_atoms + 255) / 256, batch);
    normalize_kernel<<<ng, 256, 0, stream>>>(out, maxkeys, sums, n_atoms);
}